// ScaleConv_28664611733982
// MI455X (gfx1250) — compile-verified
//
#include <hip/hip_runtime.h>
#include <hip/hip_bf16.h>

typedef __attribute__((ext_vector_type(2))) float v2f;
typedef __attribute__((ext_vector_type(8))) float v8f;

#define DFEAT 256
#define BSTRIDE 132   // 128 floats of K-half + 4 pad -> bank-conflict-free ds_load_b64

// ---------------------------------------------------------------- zero fill
__global__ __launch_bounds__(256) void zero_f32(float* __restrict__ p, long long n) {
  long long i = (long long)blockIdx.x * 256 + threadIdx.x;
  if (i < n) p[i] = 0.0f;
}

// ---------------------------------------------------------------- degrees
__global__ __launch_bounds__(256) void degree_kernel(const int* __restrict__ row,
                                                     const int* __restrict__ col,
                                                     float* __restrict__ dout,
                                                     float* __restrict__ din, int E) {
  int e = blockIdx.x * 256 + threadIdx.x;
  if (e < E) {
    unsafeAtomicAdd(dout + row[e], 1.0f);
    unsafeAtomicAdd(din + col[e], 1.0f);
  }
}

// w_a[e]  = out_deg(row)^-0.5 * in_deg(col)^-0.5   (EXPONENT = 0.5)
// w_at[e] = in_deg(col)^-0.5  * out_deg(row)^-0.5  (dir_norm_weights(col,row)) == w_a
__global__ __launch_bounds__(256) void weight_kernel(const int* __restrict__ row,
                                                     const int* __restrict__ col,
                                                     const float* __restrict__ dout,
                                                     const float* __restrict__ din,
                                                     float* __restrict__ w_a,
                                                     float* __restrict__ w_at, int E) {
  int e = blockIdx.x * 256 + threadIdx.x;
  if (e >= E) return;
  float od = dout[row[e]];
  float id = din[col[e]];
  float odr = od > 0.0f ? rsqrtf(od) : 0.0f;
  float idr = id > 0.0f ? rsqrtf(id) : 0.0f;
  float v = odr * idr;
  w_a[e] = v;
  w_at[e] = v;
}

// ---------------------------------------------------------------- SpMM (scatter-add)
// out[dst[e], :] += w[e] * h[src[e], :]   — one wave per edge, 8 floats/lane
__global__ __launch_bounds__(256) void spmm_kernel(const float* __restrict__ h,
                                                   float* __restrict__ out,
                                                   const int* __restrict__ src,
                                                   const int* __restrict__ dst,
                                                   const float* __restrict__ w, int E) {
  int e = blockIdx.x * 8 + (threadIdx.x >> 5);
  if (e >= E) return;
  const int lane = threadIdx.x & 31;
  const int s = src[e];
  const int d = dst[e];
  const float we = w[e];
  const float4* __restrict__ hs = (const float4*)(h + (size_t)s * DFEAT);
  float* __restrict__ od = out + (size_t)d * DFEAT;
#pragma unroll
  for (int i = 0; i < 2; ++i) {
    float4 v = hs[i * 32 + lane];
    float* p = od + (size_t)(i * 32 + lane) * 4;
    unsafeAtomicAdd(p + 0, we * v.x);
    unsafeAtomicAdd(p + 1, we * v.y);
    unsafeAtomicAdd(p + 2, we * v.z);
    unsafeAtomicAdd(p + 3, we * v.w);
  }
}

// ---------------------------------------------------------------- WMMA GEMM-accumulate
// out[M,256] += scale * H[M,256] @ W[256,256]^T using V_WMMA_F32_16X16X4_F32.
// Block = 8 waves; wave -> 16x64 output strip (4 accumulators reuse each A fetch).
// W rows [o0, o0+64) are staged into LDS with GLOBAL_LOAD_ASYNC_TO_LDS_B128 in two
// K-halves (64 x 132 floats, padded rows -> conflict-free ds_load_b64 fragments).
__global__ __launch_bounds__(256) void gemm_acc_wmma(const float* __restrict__ H,
                                                     const float* __restrict__ W,
                                                     float* __restrict__ out,
                                                     float scale, int mtiles) {
  __shared__ float sB[64 * BSTRIDE];

  const int lane = threadIdx.x & 31;
  const int wave = threadIdx.x >> 5;
  const int mt = blockIdx.x * 8 + wave;
  const bool active = (mt < mtiles);        // wave-uniform
  const int lmod = lane & 15;
  const int koff = (lane >> 4) << 1;        // 0 or 2
  const int o0 = blockIdx.y * 64;

  const float* __restrict__ hrow =
      H + (size_t)((active ? mt : 0) * 16 + lmod) * DFEAT + koff;
  // LDS B fragment pointers (row-major, padded stride)
  const float* __restrict__ sb0 = sB + (size_t)( 0 + lmod) * BSTRIDE + koff;
  const float* __restrict__ sb1 = sB + (size_t)(16 + lmod) * BSTRIDE + koff;
  const float* __restrict__ sb2 = sB + (size_t)(32 + lmod) * BSTRIDE + koff;
  const float* __restrict__ sb3 = sB + (size_t)(48 + lmod) * BSTRIDE + koff;

  v8f acc0 = {}; v8f acc1 = {}; v8f acc2 = {}; v8f acc3 = {};

#pragma unroll
  for (int half = 0; half < 2; ++half) {
    // ---- async-stage W[o0..o0+63][half*128 .. +128) -> LDS (all 256 threads)
    {
      const int t = threadIdx.x;
#pragma unroll
      for (int i = 0; i < 8; ++i) {
        int j = t + i * 256;                 // 16-byte chunk id, 2048 total
        int r = j >> 5;                      // W row 0..63
        int c = j & 31;                      // 16B chunk within the 128-float half
        const float* gp = W + (size_t)(o0 + r) * DFEAT + half * 128 + c * 4;
        unsigned lds = (unsigned)(size_t)(&sB[r * BSTRIDE + c * 4]);
        unsigned long long ga = (unsigned long long)(size_t)gp;
        asm volatile("global_load_async_to_lds_b128 %0, %1, off"
                     :: "v"(lds), "v"(ga) : "memory");
      }
      asm volatile("s_wait_asynccnt 0x0" ::: "memory");
    }
    __syncthreads();

    if (active) {
      const float* hk = hrow + half * 128;
#pragma unroll 4
      for (int k = 0; k < 128; k += 4) {
        v2f a  = *(const v2f*)(hk + k);
        v2f b0 = *(const v2f*)(sb0 + k);
        v2f b1 = *(const v2f*)(sb1 + k);
        v2f b2 = *(const v2f*)(sb2 + k);
        v2f b3 = *(const v2f*)(sb3 + k);
        acc0 = __builtin_amdgcn_wmma_f32_16x16x4_f32(false, a, false, b0, (short)0, acc0, false, false);
        acc1 = __builtin_amdgcn_wmma_f32_16x16x4_f32(false, a, false, b1, (short)0, acc1, false, false);
        acc2 = __builtin_amdgcn_wmma_f32_16x16x4_f32(false, a, false, b2, (short)0, acc2, false, false);
        acc3 = __builtin_amdgcn_wmma_f32_16x16x4_f32(false, a, false, b3, (short)0, acc3, false, false);
      }
    }
    __syncthreads();  // all waves done reading before next stage overwrites
  }

  if (active) {
    const int m0 = mt * 16 + (lane >> 4) * 8;
#pragma unroll
    for (int r = 0; r < 8; ++r) {
      float* p = out + (size_t)(m0 + r) * DFEAT + o0 + lmod;
      p[0]  += scale * acc0[r];
      p[16] += scale * acc1[r];
      p[32] += scale * acc2[r];
      p[48] += scale * acc3[r];
    }
  }
}

// ---------------------------------------------------------------- folded bias add
__global__ __launch_bounds__(256) void add_bias_kernel(float* __restrict__ out,
                                                       const float* __restrict__ bs2d,
                                                       const float* __restrict__ bd2s,
                                                       const float* __restrict__ bz) {
  int o = threadIdx.x;
  float b = 0.5f   * (bs2d[0 * DFEAT + o] + bd2s[0 * DFEAT + o])
          + 0.5f   *  bs2d[1 * DFEAT + o]
          + 0.25f  * (bs2d[2 * DFEAT + o] + bd2s[2 * DFEAT + o])
          + 0.25f  *  bs2d[3 * DFEAT + o]
          + 0.125f * (bs2d[4 * DFEAT + o] + bd2s[4 * DFEAT + o])
          + bz[o];
  out[(size_t)blockIdx.x * DFEAT + o] += b;
}

// ---------------------------------------------------------------- launch
extern "C" void kernel_launch(void* const* d_in, const int* in_sizes, int n_in,
                              void* d_out, int out_size, void* d_ws, size_t ws_size,
                              hipStream_t stream) {
  const float* x    = (const float*)d_in[0];
  const int*   eidx = (const int*)d_in[1];
  const float* Ws2d = (const float*)d_in[2];
  const float* bs2d = (const float*)d_in[3];
  const float* Wd2s = (const float*)d_in[4];
  const float* bd2s = (const float*)d_in[5];
  const float* Wz   = (const float*)d_in[6];
  const float* bz   = (const float*)d_in[7];
  float* out = (float*)d_out;

  const int N = in_sizes[0] / DFEAT;
  const int E = in_sizes[1] / 2;
  const int* row = eidx;       // edge_index[0]
  const int* col = eidx + E;   // edge_index[1]

  float* ws = (float*)d_ws;
  float* deg_out = ws;
  float* deg_in  = deg_out + N;
  float* w_a     = deg_in + N;
  float* w_at    = w_a + E;
  const size_t feat = (size_t)N * DFEAT;
  float* F0 = w_at + E;
  float* F1 = F0 + feat;
  float* F2 = F1 + feat;

  auto zero = [&](float* p, size_t n) {
    zero_f32<<<dim3((unsigned)((n + 255) / 256)), dim3(256), 0, stream>>>(p, (long long)n);
  };

  // degrees + normalized edge weights
  zero(deg_out, (size_t)N);
  zero(deg_in, (size_t)N);
  degree_kernel<<<dim3((E + 255) / 256), dim3(256), 0, stream>>>(row, col, deg_out, deg_in, E);
  weight_kernel<<<dim3((E + 255) / 256), dim3(256), 0, stream>>>(row, col, deg_out, deg_in,
                                                                 w_a, w_at, E);

  const int mtiles = N / 16;  // N = 50000 -> 3125
  dim3 ggrid((unsigned)((mtiles + 7) / 8), 4);
  auto gemm = [&](const float* Hm, const float* Wm, float s) {
    gemm_acc_wmma<<<ggrid, dim3(256), 0, stream>>>(Hm, Wm, out, s, mtiles);
  };
  dim3 sgrid((unsigned)((E + 7) / 8));
  auto spA = [&](const float* h, float* o) {  // o = A h : out[row] += w_a * h[col]
    zero(o, feat);
    spmm_kernel<<<sgrid, dim3(256), 0, stream>>>(h, o, col, row, w_a, E);
  };
  auto spAt = [&](const float* h, float* o) { // o = At h : out[col] += w_at * h[row]
    zero(o, feat);
    spmm_kernel<<<sgrid, dim3(256), 0, stream>>>(h, o, row, col, w_at, E);
  };

  const size_t WSZ = (size_t)DFEAT * DFEAT;  // 65536 elements per weight matrix

  zero(out, feat);
  // ---- chains seeded by Y = A x
  spA(x, F0);    gemm(F0, Ws2d + 0 * WSZ, 0.5f);    // y         * alpha
  spA(F0, F1);   gemm(F1, Ws2d + 2 * WSZ, 0.25f);   // yy(i=1)   * gamma*w1
  spA(F1, F2);   gemm(F2, Ws2d + 4 * WSZ, 0.125f);  // yy(i=2)   * gamma*w2
  spAt(F0, F1);  gemm(F1, Ws2d + 1 * WSZ, 0.25f);   // yty(i=1)  * (1-beta)*w1
  spAt(F1, F2);  gemm(F2, Ws2d + 3 * WSZ, 0.125f);  // yty(i=2)  * (1-beta)*w2
  // ---- chains seeded by YT = At x
  spAt(x, F0);   gemm(F0, Wd2s + 0 * WSZ, 0.5f);    // y_t       * (1-alpha)
  spA(F0, F1);   gemm(F1, Ws2d + 1 * WSZ, 0.25f);   // yyt(i=1)  * beta*w1
  spA(F1, F2);   gemm(F2, Ws2d + 3 * WSZ, 0.125f);  // yyt(i=2)  * beta*w2
  spAt(F0, F1);  gemm(F1, Wd2s + 2 * WSZ, 0.25f);   // ytyt(i=1) * (1-gamma)*w1
  spAt(F1, F2);  gemm(F2, Wd2s + 4 * WSZ, 0.125f);  // ytyt(i=2) * (1-gamma)*w2
  // ---- zero order
  gemm(x, Wz, 1.0f);
  // ---- folded biases
  add_bias_kernel<<<dim3((unsigned)N), dim3(256), 0, stream>>>(out, bs2d, bd2s, bz);
}